// UPPST_69544110457224
// MI455X (gfx1250) — compile-verified
//
#include <hip/hip_runtime.h>
#include <math.h>

typedef __attribute__((ext_vector_type(2))) float v2f;
typedef __attribute__((ext_vector_type(8))) float v8f;

#define HDIM 128

__device__ __forceinline__ void atomAddF(float* p, float v) {
    __hip_atomic_fetch_add(p, v, __ATOMIC_RELAXED, __HIP_MEMORY_SCOPE_AGENT);
}

// ---------- GCN normalization ----------
__global__ __launch_bounds__(256) void k_fill1(float* __restrict__ p, int n) {
    int i = blockIdx.x * 256 + threadIdx.x;
    if (i < n) p[i] = 1.0f;              // self-loop weight 1 seeds the degree
}

__global__ __launch_bounds__(256) void k_degacc(float* __restrict__ deg,
                                                const int* __restrict__ dst,
                                                const float* __restrict__ ew, int e) {
    int i = blockIdx.x * 256 + threadIdx.x;
    if (i < e) atomAddF(&deg[dst[i]], ew[i]);
}

__global__ __launch_bounds__(256) void k_rsqrt(float* __restrict__ p, int n) {
    int i = blockIdx.x * 256 + threadIdx.x;
    if (i < n) { float d = p[i]; p[i] = d > 0.0f ? rsqrtf(d) : 0.0f; }
}

__global__ __launch_bounds__(256) void k_norm(float* __restrict__ nrm,
                                              const float* __restrict__ dinv,
                                              const int* __restrict__ src,
                                              const int* __restrict__ dst,
                                              const float* __restrict__ ew, int e) {
    int i = blockIdx.x * 256 + threadIdx.x;
    if (i < e) nrm[i] = dinv[src[i]] * ew[i] * dinv[dst[i]];
}

// ---------- dense GEMM: C[N,128] = A[N,128] @ W[128,128] via fp32 WMMA ----------
// 256 threads = 8 waves; W staged in 64KB LDS and shared by the 8 waves.
// Each wave computes one 16-row stripe: 8 x (16x16) tiles, K=128 in 32 k-steps.
__global__ __launch_bounds__(256) void k_gemm128(const float* __restrict__ A,
                                                 const float* __restrict__ W,
                                                 float* __restrict__ C,
                                                 const int* __restrict__ perm,
                                                 int nrows) {
    __shared__ float sW[HDIM * HDIM];
    for (int i = threadIdx.x * 4; i < HDIM * HDIM; i += 256 * 4) {
        float4 t = *(const float4*)(W + i);
        *(float4*)(sW + i) = t;
    }
    __syncthreads();

    int wave = threadIdx.x >> 5;
    int lane = threadIdx.x & 31;
    int stripe = blockIdx.x * 8 + wave;
    int nstripes = nrows >> 4;               // nrows % 16 == 0
    if (stripe >= nstripes) return;          // wave-uniform: EXEC stays all-ones

    int m0 = stripe * 16;
    int hl = lane >> 4;                      // half: 0 -> K=0,1 ; 1 -> K=2,3
    int ln = lane & 15;
    int arow = m0 + ln;
    if (perm) arow = perm[arow];             // row gather for corrupted encoder
    const float* Arow = A + (size_t)arow * HDIM;

    v8f acc[8] = {};
#pragma unroll 4
    for (int k0 = 0; k0 < HDIM; k0 += 4) {
        v2f a;
        a.x = Arow[k0 + hl * 2];
        a.y = Arow[k0 + hl * 2 + 1];
        const float* bptr = sW + (k0 + hl * 2) * HDIM + ln;
#pragma unroll
        for (int t = 0; t < 8; t++) {
            v2f b;
            b.x = bptr[t * 16];              // W[k  ][n0+ln]
            b.y = bptr[t * 16 + HDIM];       // W[k+1][n0+ln]
            acc[t] = __builtin_amdgcn_wmma_f32_16x16x4_f32(
                false, a, false, b, (short)0, acc[t], false, false);
        }
    }

    float* Cr = C + (size_t)m0 * HDIM;
#pragma unroll
    for (int t = 0; t < 8; t++)
#pragma unroll
        for (int v = 0; v < 8; v++)
            Cr[(size_t)(v + hl * 8) * HDIM + t * 16 + ln] = acc[t][v];
}

// ---------- propagation: out = b + selfnorm*h  then  out[dst] += norm*h[src] ----------
__global__ __launch_bounds__(256) void k_prop_init(const float* __restrict__ h,
                                                   float* __restrict__ o,
                                                   const float* __restrict__ bias,
                                                   const float* __restrict__ dinv, int n) {
    int wid = (blockIdx.x * 256 + threadIdx.x) >> 5;
    int lane = threadIdx.x & 31;
    if (wid >= n) return;
    float di = dinv[wid];
    float sn = di * di;                      // self-loop norm
    float4 hv = *(const float4*)(h + (size_t)wid * HDIM + lane * 4);
    float4 bv = *(const float4*)(bias + lane * 4);
    float4 ov = {bv.x + hv.x * sn, bv.y + hv.y * sn,
                 bv.z + hv.z * sn, bv.w + hv.w * sn};
    *(float4*)(o + (size_t)wid * HDIM + lane * 4) = ov;
}

__global__ __launch_bounds__(256) void k_prop_edges(const float* __restrict__ h,
                                                    float* __restrict__ o,
                                                    const int* __restrict__ src,
                                                    const int* __restrict__ dst,
                                                    const float* __restrict__ nrm, int e) {
    int wid = (blockIdx.x * 256 + threadIdx.x) >> 5;  // one wave per edge
    int lane = threadIdx.x & 31;
    if (wid >= e) return;
    int s = src[wid], d = dst[wid];
    float w = nrm[wid];
    float4 hv = *(const float4*)(h + (size_t)s * HDIM + lane * 4);
    float* op = o + (size_t)d * HDIM + lane * 4;
    atomAddF(op + 0, hv.x * w);
    atomAddF(op + 1, hv.y * w);
    atomAddF(op + 2, hv.z * w);
    atomAddF(op + 3, hv.w * w);
}

__global__ __launch_bounds__(256) void k_prelu(float* __restrict__ z,
                                               const float* __restrict__ a, long long n) {
    long long i = (long long)blockIdx.x * 256 + threadIdx.x;
    if (i < n) {
        float v = z[i];
        z[i] = v >= 0.0f ? v : a[(int)(i & (HDIM - 1))] * v;
    }
}

// ---------- summary = sigmoid(mean over rows) ----------
__global__ void k_zero128(float* __restrict__ p) { p[threadIdx.x] = 0.0f; }

__global__ __launch_bounds__(128) void k_colsum(const float* __restrict__ z,
                                                float* __restrict__ acc, int n) {
    int c = threadIdx.x;
    int r0 = blockIdx.x * 256;
    int r1 = r0 + 256; if (r1 > n) r1 = n;
    float s = 0.0f;
    for (int r = r0; r < r1; r++) s += z[(size_t)r * HDIM + c];
    atomAddF(&acc[c], s);
}

__global__ void k_summary(const float* __restrict__ acc, float* __restrict__ sm, int n) {
    int c = threadIdx.x;
    sm[c] = 1.0f / (1.0f + expf(-acc[c] / (float)n));
}

// ---------- Student's-t soft assignment: one wave per node ----------
__global__ __launch_bounds__(256) void k_q(const float* __restrict__ xo,
                                           const float* __restrict__ mu,
                                           float* __restrict__ q, int n, int K) {
    int wid = (blockIdx.x * 256 + threadIdx.x) >> 5;
    int lane = threadIdx.x & 31;
    if (wid >= n) return;
    float4 xv = *(const float4*)(xo + (size_t)wid * HDIM + lane * 4);
    float xx = xv.x * xv.x + xv.y * xv.y + xv.z * xv.z + xv.w * xv.w;
    for (int o = 16; o > 0; o >>= 1) xx += __shfl_xor(xx, o, 32);

    float myq = 0.0f, qsum = 0.0f;
    for (int k = 0; k < K; k++) {
        float4 mv = *(const float4*)(mu + (size_t)k * HDIM + lane * 4);
        float dot = xv.x * mv.x + xv.y * mv.y + xv.z * mv.z + xv.w * mv.w;
        float mm  = mv.x * mv.x + mv.y * mv.y + mv.z * mv.z + mv.w * mv.w;
        for (int o = 16; o > 0; o >>= 1) {
            dot += __shfl_xor(dot, o, 32);
            mm  += __shfl_xor(mm, o, 32);
        }
        float d = xx + mm - 2.0f * dot;
        d = fmaxf(d, 0.0f);
        float qq = 1.0f / (1.0f + d * 5.0f + 1e-8f);     // d/ALPHA, ALPHA=0.2
        qq = __powf(qq, 1.2f) * 0.5f;                    // q^(ALPHA+1) / 2
        qsum += qq;
        if (lane == k) myq = qq;
    }
    if (lane < K) q[(size_t)wid * K + lane] = myq / qsum;
}

// ---------- driver ----------
extern "C" void kernel_launch(void* const* d_in, const int* in_sizes, int n_in,
                              void* d_out, int out_size, void* d_ws, size_t ws_size,
                              hipStream_t stream) {
    const float* x       = (const float*)d_in[0];
    const int*   ei      = (const int*)d_in[1];
    const float* ew      = (const float*)d_in[2];
    const int*   perm    = (const int*)d_in[3];
    const float* W1      = (const float*)d_in[4];
    const float* b1      = (const float*)d_in[5];
    const float* W2      = (const float*)d_in[6];
    const float* b2      = (const float*)d_in[7];
    const float* W3      = (const float*)d_in[8];
    const float* b3      = (const float*)d_in[9];
    const float* W4      = (const float*)d_in[10];
    const float* b4      = (const float*)d_in[11];
    const float* prelu_a = (const float*)d_in[12];
    const float* Wc      = (const float*)d_in[13];
    const float* bc      = (const float*)d_in[14];
    const float* mu      = (const float*)d_in[15];

    int N = in_sizes[0] / HDIM;
    int E = in_sizes[1] / 2;
    int K = in_sizes[15] / HDIM;
    const int* src = ei;
    const int* dst = ei + E;

    float* ws   = (float*)d_ws;
    float* dinv = ws;                            // [N]
    float* nrm  = dinv + N;                      // [E]
    float* g    = nrm + E;                       // [N,128] gemm out
    float* f    = g + (size_t)N * HDIM;          // [N,128] feature ping
    float* acc  = f + (size_t)N * HDIM;          // [128]

    float* outp = (float*)d_out;
    float* pos  = outp;
    float* neg  = pos + (size_t)N * HDIM;
    float* summ = neg + (size_t)N * HDIM;
    float* xo   = summ + HDIM;
    float* q    = xo + (size_t)N * HDIM;

    int gN   = (N + 255) / 256;
    int gE   = (E + 255) / 256;
    int gGem = (N / 16 + 7) / 8;                 // 8 stripes (waves) per block
    int gNW  = (N + 7) / 8;                      // one wave per node
    int gEW  = (E + 7) / 8;                      // one wave per edge
    long long nElem = (long long)N * HDIM;
    int gElem = (int)((nElem + 255) / 256);

    // gcn_norm
    k_fill1 <<<gN, 256, 0, stream>>>(dinv, N);
    k_degacc<<<gE, 256, 0, stream>>>(dinv, dst, ew, E);
    k_rsqrt <<<gN, 256, 0, stream>>>(dinv, N);
    k_norm  <<<gE, 256, 0, stream>>>(nrm, dinv, src, dst, ew, E);

    const float* Ws[4] = {W1, W2, W3, W4};
    const float* bs[4] = {b1, b2, b3, b4};

    // positive encoder
    const float* cur = x;
    for (int l = 0; l < 4; l++) {
        float* dbuf = (l == 3) ? pos : f;
        k_gemm128   <<<gGem, 256, 0, stream>>>(cur, Ws[l], g, nullptr, N);
        k_prop_init <<<gNW,  256, 0, stream>>>(g, dbuf, bs[l], dinv, N);
        k_prop_edges<<<gEW,  256, 0, stream>>>(g, dbuf, src, dst, nrm, E);
        cur = dbuf;
    }
    k_prelu<<<gElem, 256, 0, stream>>>(pos, prelu_a, nElem);

    // corrupted encoder (row permutation fused into first GEMM's row gather)
    cur = x;
    for (int l = 0; l < 4; l++) {
        float* dbuf = (l == 3) ? neg : f;
        k_gemm128   <<<gGem, 256, 0, stream>>>(cur, Ws[l], g, (l == 0) ? perm : nullptr, N);
        k_prop_init <<<gNW,  256, 0, stream>>>(g, dbuf, bs[l], dinv, N);
        k_prop_edges<<<gEW,  256, 0, stream>>>(g, dbuf, src, dst, nrm, E);
        cur = dbuf;
    }
    k_prelu<<<gElem, 256, 0, stream>>>(neg, prelu_a, nElem);

    // summary = sigmoid(mean(pos_z, axis=0))
    k_zero128<<<1, 128, 0, stream>>>(acc);
    k_colsum <<<(N + 255) / 256, 128, 0, stream>>>(pos, acc, N);
    k_summary<<<1, 128, 0, stream>>>(acc, summ, N);

    // decoder conv: x_out = gcn_conv(pos_z, Wc, bc)
    k_gemm128   <<<gGem, 256, 0, stream>>>(pos, Wc, g, nullptr, N);
    k_prop_init <<<gNW,  256, 0, stream>>>(g, xo, bc, dinv, N);
    k_prop_edges<<<gEW,  256, 0, stream>>>(g, xo, src, dst, nrm, E);

    // soft assignment
    k_q<<<gNW, 256, 0, stream>>>(xo, mu, q, N, K);

    (void)n_in; (void)out_size; (void)ws_size;
}